// Decoder_11519102288461
// MI455X (gfx1250) — compile-verified
//
#include <hip/hip_runtime.h>
#include <math.h>

#define V 50257
#define H 1024
#define L 20

typedef __attribute__((ext_vector_type(2))) float v2f;
typedef __attribute__((ext_vector_type(8))) float v8f;

// ---------------------------------------------------------------------------
// K1: embedding + attention softmax + focused = attn_w @ encoder_outputs
// single block, 256 threads (8 wave32s)
// ---------------------------------------------------------------------------
__global__ void k_attention(const int* __restrict__ prev_word,
                            const float* __restrict__ prev_hidden,
                            const float* __restrict__ enc,
                            const float* __restrict__ emb,
                            const float* __restrict__ attn_W,
                            const float* __restrict__ attn_b,
                            float* __restrict__ focused) {
  __shared__ float sm[L];
  __shared__ float aw[L];
  const int t = threadIdx.x;
  const int lane = t & 31;
  const int wave = t >> 5;
  const int w = prev_word[0];
  const float* erow = emb + (size_t)w * H;

  // attention logits: wave handles rows j = wave, wave+8, wave+16
  for (int j = wave; j < L; j += 8) {
    const float* wr = attn_W + (size_t)j * (2 * H);
    float acc = 0.f;
    for (int c = lane; c < H; c += 32) acc += erow[c] * wr[c];
    for (int c = lane; c < H; c += 32) acc += prev_hidden[c] * wr[H + c];
    for (int off = 16; off > 0; off >>= 1) acc += __shfl_xor(acc, off, 32);
    if (lane == 0) sm[j] = acc + attn_b[j];
  }
  __syncthreads();
  if (t == 0) {  // softmax over 20 values: trivial, one thread
    float m = sm[0];
    for (int j = 1; j < L; ++j) m = fmaxf(m, sm[j]);
    float s = 0.f;
    for (int j = 0; j < L; ++j) { aw[j] = expf(sm[j] - m); s += aw[j]; }
    float inv = 1.f / s;
    for (int j = 0; j < L; ++j) aw[j] *= inv;
  }
  __syncthreads();
  for (int i = t; i < H; i += 256) {
    float acc = 0.f;
    for (int j = 0; j < L; ++j) acc += aw[j] * enc[j * H + i];
    focused[i] = acc;
  }
}

// ---------------------------------------------------------------------------
// K2: gru_in = relu([focused, embedded] @ comb_W.T + comb_b)   (one wave/row)
// grid: 128 blocks x 256 threads -> 1024 waves
// ---------------------------------------------------------------------------
__global__ void k_combine(const int* __restrict__ prev_word,
                          const float* __restrict__ emb,
                          const float* __restrict__ focused,
                          const float* __restrict__ comb_W,
                          const float* __restrict__ comb_b,
                          float* __restrict__ gru_in) {
  const int t = threadIdx.x;
  const int lane = t & 31;
  const int j = blockIdx.x * 8 + (t >> 5);  // 0..1023
  const int w = prev_word[0];
  const float* erow = emb + (size_t)w * H;
  const float* wr = comb_W + (size_t)j * (2 * H);
  float acc = 0.f;
  for (int c = lane; c < H; c += 32) acc += focused[c] * wr[c];
  for (int c = lane; c < H; c += 32) acc += erow[c] * wr[H + c];
  for (int off = 16; off > 0; off >>= 1) acc += __shfl_xor(acc, off, 32);
  if (lane == 0) gru_in[j] = fmaxf(acc + comb_b[j], 0.f);
}

// ---------------------------------------------------------------------------
// K3: full GRU cell, one wave per hidden unit j (6 dot-1024s per wave)
// grid: 128 blocks x 256 threads
// ---------------------------------------------------------------------------
__global__ void k_gru(const float* __restrict__ prev_hidden,
                      const float* __restrict__ gru_in,
                      const float* __restrict__ w_ih,
                      const float* __restrict__ w_hh,
                      const float* __restrict__ b_ih,
                      const float* __restrict__ b_hh,
                      float* __restrict__ hnew,
                      float* __restrict__ out_h) {
  const int t = threadIdx.x;
  const int lane = t & 31;
  const int j = blockIdx.x * 8 + (t >> 5);  // 0..1023
  const float* wi_r = w_ih + (size_t)j * H;
  const float* wi_z = w_ih + (size_t)(j + H) * H;
  const float* wi_n = w_ih + (size_t)(j + 2 * H) * H;
  const float* wh_r = w_hh + (size_t)j * H;
  const float* wh_z = w_hh + (size_t)(j + H) * H;
  const float* wh_n = w_hh + (size_t)(j + 2 * H) * H;
  float ir = 0.f, iz = 0.f, inn = 0.f, hr = 0.f, hz = 0.f, hn = 0.f;
  for (int c = lane; c < H; c += 32) {
    const float gi = gru_in[c];
    const float hh = prev_hidden[c];
    ir += gi * wi_r[c]; iz += gi * wi_z[c]; inn += gi * wi_n[c];
    hr += hh * wh_r[c]; hz += hh * wh_z[c]; hn += hh * wh_n[c];
  }
  for (int off = 16; off > 0; off >>= 1) {
    ir += __shfl_xor(ir, off, 32);  iz += __shfl_xor(iz, off, 32);
    inn += __shfl_xor(inn, off, 32); hr += __shfl_xor(hr, off, 32);
    hz += __shfl_xor(hz, off, 32);  hn += __shfl_xor(hn, off, 32);
  }
  if (lane == 0) {
    ir += b_ih[j];          hr += b_hh[j];
    iz += b_ih[j + H];      hz += b_hh[j + H];
    inn += b_ih[j + 2 * H]; hn += b_hh[j + 2 * H];
    const float r = 1.f / (1.f + expf(-(ir + hr)));
    const float z = 1.f / (1.f + expf(-(iz + hz)));
    const float n = tanhf(inn + r * hn);
    const float hv = (1.f - z) * n + z * prev_hidden[j];
    hnew[j] = hv;
    out_h[j] = hv;
  }
}

// ---------------------------------------------------------------------------
// K4: logits = h_new @ out_W.T + out_b via V_WMMA_F32_16X16X4_F32.
// One wave per 16-row vocab tile. A = x-chunk broadcast over M (documented
// layout: lane<16 -> K=0,1 ; lane>=16 -> K=2,3). B = 16x4 slice of out_W^T
// (row-striped-across-lanes, mirrored half layout like C/D). All rows of D
// are then identical: D[.][n] = sum_k x[k]*W[row0+n][k].
// ---------------------------------------------------------------------------
__global__ void k_logits_wmma(const float* __restrict__ x,
                              const float* __restrict__ out_W,
                              const float* __restrict__ out_b,
                              float* __restrict__ logits) {
  __shared__ float xs[H];
  const int t = threadIdx.x;
  for (int i = t; i < H; i += 256) xs[i] = x[i];
  __syncthreads();

  const int lane = t & 31;
  const int tile = blockIdx.x * 8 + (t >> 5);
  constexpr int NTILES = (V + 15) / 16;  // 3142
  if (tile >= NTILES) return;            // wave-uniform: EXEC stays all-ones

  const int row0 = tile * 16;
  int r = row0 + (lane & 15);
  if (r >= V) r = V - 1;                 // tail tile: duplicate last row
  const int hi = (lane >= 16) ? 2 : 0;   // K sub-offset held by this half-wave
  const float* wrow = out_W + (size_t)r * H + hi;

  v8f c0 = {}; v8f c1 = {}; v8f c2 = {}; v8f c3 = {};
  for (int k0 = 0; k0 < H; k0 += 16) {
    const v2f a0 = { xs[k0 + hi],      xs[k0 + hi + 1] };
    const v2f a1 = { xs[k0 + 4 + hi],  xs[k0 + 5 + hi] };
    const v2f a2 = { xs[k0 + 8 + hi],  xs[k0 + 9 + hi] };
    const v2f a3 = { xs[k0 + 12 + hi], xs[k0 + 13 + hi] };
    const v2f b0 = *(const v2f*)(wrow + k0);
    const v2f b1 = *(const v2f*)(wrow + k0 + 4);
    const v2f b2 = *(const v2f*)(wrow + k0 + 8);
    const v2f b3 = *(const v2f*)(wrow + k0 + 12);
    c0 = __builtin_amdgcn_wmma_f32_16x16x4_f32(false, a0, false, b0, (short)0, c0, false, false);
    c1 = __builtin_amdgcn_wmma_f32_16x16x4_f32(false, a1, false, b1, (short)0, c1, false, false);
    c2 = __builtin_amdgcn_wmma_f32_16x16x4_f32(false, a2, false, b2, (short)0, c2, false, false);
    c3 = __builtin_amdgcn_wmma_f32_16x16x4_f32(false, a3, false, b3, (short)0, c3, false, false);
  }
  const v8f cs = c0 + c1 + c2 + c3;
  // every D row is identical; lane n (0..15) holds D[0][n] in element 0
  if (lane < 16) {
    const int rr = row0 + lane;
    if (rr < V) logits[rr] = cs[0] + out_b[rr];
  }
}

// ---------------------------------------------------------------------------
// K5a/K5b: two-stage online logsumexp over V logits
// ---------------------------------------------------------------------------
__global__ void k_lse_partial(const float* __restrict__ logits,
                              float* __restrict__ pm, float* __restrict__ ps) {
  const int t = threadIdx.x;
  float m = -INFINITY, s = 0.f;
  for (int i = blockIdx.x * 256 + t; i < V; i += 64 * 256) {
    const float v = logits[i];
    if (v > m) { s = s * expf(m - v) + 1.f; m = v; }
    else       { s += expf(v - m); }
  }
  __shared__ float smv[256], ssv[256];
  smv[t] = m; ssv[t] = s;
  __syncthreads();
  for (int off = 128; off > 0; off >>= 1) {
    if (t < off) {
      const float m1 = smv[t], s1 = ssv[t];
      const float m2 = smv[t + off], s2 = ssv[t + off];
      const float M = fmaxf(m1, m2);
      float S = 0.f;
      if (s1 > 0.f) S += s1 * expf(m1 - M);
      if (s2 > 0.f) S += s2 * expf(m2 - M);
      smv[t] = M; ssv[t] = S;
    }
    __syncthreads();
  }
  if (t == 0) { pm[blockIdx.x] = smv[0]; ps[blockIdx.x] = ssv[0]; }
}

__global__ void k_lse_final(const float* __restrict__ pm,
                            const float* __restrict__ ps,
                            float* __restrict__ lse) {
  if (threadIdx.x == 0) {
    float M = -INFINITY;
    for (int i = 0; i < 64; ++i) M = fmaxf(M, pm[i]);
    float S = 0.f;
    for (int i = 0; i < 64; ++i)
      if (ps[i] > 0.f) S += ps[i] * expf(pm[i] - M);
    lse[0] = M + logf(S);
  }
}

// ---------------------------------------------------------------------------
// K6: probs = logits - lse
// ---------------------------------------------------------------------------
__global__ void k_writeout(const float* __restrict__ logits,
                           const float* __restrict__ lse,
                           float* __restrict__ out) {
  const int i = blockIdx.x * 256 + threadIdx.x;
  if (i < V) out[i] = logits[i] - lse[0];
}

// ---------------------------------------------------------------------------
extern "C" void kernel_launch(void* const* d_in, const int* in_sizes, int n_in,
                              void* d_out, int out_size, void* d_ws, size_t ws_size,
                              hipStream_t stream) {
  const int*   prev_word   = (const int*)  d_in[0];
  const float* prev_hidden = (const float*)d_in[1];
  const float* enc         = (const float*)d_in[2];
  const float* emb         = (const float*)d_in[3];
  const float* attn_W      = (const float*)d_in[4];
  const float* attn_b      = (const float*)d_in[5];
  const float* comb_W      = (const float*)d_in[6];
  const float* comb_b      = (const float*)d_in[7];
  const float* w_ih        = (const float*)d_in[8];
  const float* w_hh        = (const float*)d_in[9];
  const float* b_ih        = (const float*)d_in[10];
  const float* b_hh        = (const float*)d_in[11];
  const float* out_W       = (const float*)d_in[12];
  const float* out_b       = (const float*)d_in[13];
  float* out = (float*)d_out;  // [probs(V) | h_new(H)]

  float* ws      = (float*)d_ws;
  float* focused = ws;                   // H
  float* gru_in  = ws + 1024;            // H
  float* hnew    = ws + 2048;            // H
  float* logits  = ws + 3072;            // V (padded to 50272)
  float* pm      = ws + 3072 + 50272;    // 64
  float* ps      = pm + 64;              // 64
  float* lse     = ps + 64;              // 1

  constexpr int NTILES = (V + 15) / 16;          // 3142
  constexpr int WMMA_BLOCKS = (NTILES + 7) / 8;  // 393

  k_attention<<<1, 256, 0, stream>>>(prev_word, prev_hidden, enc, emb,
                                     attn_W, attn_b, focused);
  k_combine<<<128, 256, 0, stream>>>(prev_word, emb, focused, comb_W, comb_b,
                                     gru_in);
  k_gru<<<128, 256, 0, stream>>>(prev_hidden, gru_in, w_ih, w_hh, b_ih, b_hh,
                                 hnew, out + V);
  k_logits_wmma<<<WMMA_BLOCKS, 256, 0, stream>>>(hnew, out_W, out_b, logits);
  k_lse_partial<<<64, 256, 0, stream>>>(logits, pm, ps);
  k_lse_final<<<1, 64, 0, stream>>>(pm, ps, lse);
  k_writeout<<<(V + 255) / 256, 256, 0, stream>>>(logits, lse, out);
}